// ActSWL_80058190397923
// MI455X (gfx1250) — compile-verified
//
#include <hip/hip_runtime.h>

// ActSWL: X_t = cumsum_t(x), Y_t = silu(X_t), out_t = Y_t - Y_{t-1}
// T = 4 (unrolled), slice N = B*L*D elements. Bandwidth-bound streaming:
// 256 MB in + 256 MB out -> ~22 us floor at 23.3 TB/s. Total tensor is
// 256 MB so ALL byte offsets fit in signed 32-bit -> use 32-bit indexing
// so the compiler emits GVS addressing (SGPR64 base + VGPR32 offset).

typedef __attribute__((ext_vector_type(4))) float f4;

__device__ __forceinline__ float silu1(float v) {
    // v * sigmoid(v) = v * rcp(1 + exp(-v)); v_exp_f32 + v_rcp_f32 fast path
    return v * __builtin_amdgcn_rcpf(1.0f + __expf(-v));
}

__device__ __forceinline__ f4 silu4(f4 v) {
    f4 r;
    r.x = silu1(v.x);
    r.y = silu1(v.y);
    r.z = silu1(v.z);
    r.w = silu1(v.w);
    return r;
}

__global__ __launch_bounds__(256) void ActSWL_kernel(const float* __restrict__ x,
                                                     float* __restrict__ out,
                                                     unsigned n4,      // float4 chunks per t-slice
                                                     unsigned nslice)  // elements per t-slice
{
    const unsigned s4 = nslice >> 2;   // float4 stride between t-slices
    const f4* __restrict__ xi = (const f4*)x;
    f4* __restrict__ oo = (f4*)out;

    unsigned tid    = blockIdx.x * blockDim.x + threadIdx.x;
    unsigned stride = gridDim.x * blockDim.x;

    for (unsigned i = tid; i < n4; i += stride) {
        // Prefetch next grid-stride iteration (gfx1250: global_prefetch_b8).
        unsigned nx = i + stride;
        if (nx < n4) {
            __builtin_prefetch(&xi[nx], 0, 0);
            __builtin_prefetch(&xi[nx + s4], 0, 0);
            __builtin_prefetch(&xi[nx + 2u * s4], 0, 0);
            __builtin_prefetch(&xi[nx + 3u * s4], 0, 0);
        }

        // Stream in all 4 time steps for this (b,l,d) chunk. NT hints:
        // no reuse, 512 MB total traffic >> 192 MB L2.
        f4 a0 = __builtin_nontemporal_load(&xi[i]);
        f4 a1 = __builtin_nontemporal_load(&xi[i + s4]);
        f4 a2 = __builtin_nontemporal_load(&xi[i + 2u * s4]);
        f4 a3 = __builtin_nontemporal_load(&xi[i + 3u * s4]);

        // Cumsum along T in registers.
        f4 X0 = a0;
        f4 X1 = X0 + a1;
        f4 X2 = X1 + a2;
        f4 X3 = X2 + a3;

        f4 Y0 = silu4(X0);
        f4 Y1 = silu4(X1);
        f4 Y2 = silu4(X2);
        f4 Y3 = silu4(X3);

        f4 o0 = Y0;
        f4 o1 = Y1 - Y0;
        f4 o2 = Y2 - Y1;
        f4 o3 = Y3 - Y2;

        __builtin_nontemporal_store(o0, &oo[i]);
        __builtin_nontemporal_store(o1, &oo[i + s4]);
        __builtin_nontemporal_store(o2, &oo[i + 2u * s4]);
        __builtin_nontemporal_store(o3, &oo[i + 3u * s4]);
    }

    // Scalar tail (nslice % 4 != 0); no-op for the reference shape (D=4096).
    unsigned tail_base = n4 << 2;
    unsigned rem = nslice - tail_base;
    if (tid < rem) {
        unsigned e = tail_base + tid;
        float b0 = x[e];
        float b1 = x[e + nslice];
        float b2 = x[e + 2u * nslice];
        float b3 = x[e + 3u * nslice];
        float X0 = b0, X1 = X0 + b1, X2 = X1 + b2, X3 = X2 + b3;
        float Y0 = silu1(X0), Y1 = silu1(X1), Y2 = silu1(X2), Y3 = silu1(X3);
        out[e]               = Y0;
        out[e + nslice]      = Y1 - Y0;
        out[e + 2u * nslice] = Y2 - Y1;
        out[e + 3u * nslice] = Y3 - Y2;
    }
}

extern "C" void kernel_launch(void* const* d_in, const int* in_sizes, int n_in,
                              void* d_out, int out_size, void* d_ws, size_t ws_size,
                              hipStream_t stream) {
    (void)n_in; (void)d_ws; (void)ws_size; (void)out_size;

    const float* x = (const float*)d_in[0];
    float* out = (float*)d_out;

    const unsigned T = 4;
    unsigned total  = (unsigned)in_sizes[0];  // T*B*L*D = 67,108,864
    unsigned nslice = total / T;              // B*L*D   = 16,777,216
    unsigned n4     = nslice >> 2;            // float4 chunks = 4,194,304

    const int threads = 256;
    unsigned want = (n4 + threads - 1) / threads;
    // Cap grid so each thread runs ~2 grid-stride iterations: the prefetch of
    // iteration i+1 overlaps the 16 exp/rcp chains of iteration i.
    unsigned cap = 8192;
    unsigned blocks = want < cap ? (want < 1 ? 1u : want) : cap;

    ActSWL_kernel<<<blocks, threads, 0, stream>>>(x, out, n4, nslice);
}